// DescriptionEmbedding_6167573037096
// MI455X (gfx1250) — compile-verified
//
#include <hip/hip_runtime.h>
#include <hip/hip_bf16.h>
#include <math.h>

typedef __attribute__((ext_vector_type(2))) float v2f;
typedef __attribute__((ext_vector_type(8))) float v8f;

#if __has_builtin(__builtin_amdgcn_tanhf)
#define FAST_TANH(x) __builtin_amdgcn_tanhf(x)
#else
#define FAST_TANH(x) tanhf(x)
#endif

// ---------------------------------------------------------------------------
// Kernel A: out[M,64] = A[M,16] @ Wm[16,64] (+ bias[64])
// One wave (32 threads) per (16-row M-tile, 16-col N-tile); 4 N-tiles per row.
// Uses V_WMMA_F32_16X16X4_F32, K = 16 -> 4 WMMA steps.
// f32 A-operand layout (16x4): lanes 0-15 M=0..15 {K=kk,kk+1}, lanes 16-31 {K=kk+2,kk+3}.
// ---------------------------------------------------------------------------
__global__ void wmma_proj_kernel(const float* __restrict__ A,
                                 const float* __restrict__ Wm,
                                 const float* __restrict__ bias,
                                 float* __restrict__ out,
                                 int M)
{
    const int tile = blockIdx.x;
    const int mt   = tile >> 2;     // M tile index
    const int nt   = tile & 3;      // N tile index (N=64 -> 4 tiles)
    const int lane = threadIdx.x;
    const int half = lane >> 4;     // 0: K pair {0,1}, 1: K pair {2,3}
    const int l16  = lane & 15;
    const int row  = mt * 16 + l16; // A row for this lane
    const int ncol = nt * 16 + l16; // B/D column for this lane

    v8f acc = {};
#pragma unroll
    for (int kk = 0; kk < 16; kk += 4) {
        const int k = kk + half * 2;
        v2f a, b;
        a.x = A[(size_t)row * 16 + k];
        a.y = A[(size_t)row * 16 + k + 1];
        b.x = Wm[(size_t)k * 64 + ncol];
        b.y = Wm[(size_t)(k + 1) * 64 + ncol];
        acc = __builtin_amdgcn_wmma_f32_16x16x4_f32(
            /*neg_a=*/false, a, /*neg_b=*/false, b,
            /*c_mod=*/(short)0, acc, /*reuse_a=*/false, /*reuse_b=*/false);
    }

    const float bv = bias ? bias[ncol] : 0.0f;
    // D layout: VGPR r, lanes 0-15 -> M=r, lanes 16-31 -> M=r+8; N = lane&15
#pragma unroll
    for (int r = 0; r < 8; ++r) {
        const int mrow = mt * 16 + r + half * 8;
        out[(size_t)mrow * 64 + ncol] = acc[r] + bv;
    }
}

// ---------------------------------------------------------------------------
// Kernel B (dominant): fused tanh-MLP score + masked softmax over W + context.
// One block per f; 256 threads stride over W (4000). hid_part is 1 MB and
// L2-resident; feat row + W_u broadcast from LDS. ~16 w-iters/thread,
// 64 v_tanh_f32 + 64 v_fmac per w. Reduction: wave32 shfl_xor tree, then
// ds_add_f32 LDS atomics, then 1 divide per context element.
// ---------------------------------------------------------------------------
__global__ void attn_context_kernel(const float* __restrict__ feat_part,  // [F,64]
                                    const float* __restrict__ hid_part,   // [W,64] (b_w folded)
                                    const float* __restrict__ W_u,        // [64]
                                    const float* __restrict__ hidden_emb, // [W,16]
                                    const unsigned char* __restrict__ mask, // [F,W] bool
                                    float* __restrict__ context,          // [F,16]
                                    int W)
{
    __shared__ float sAf[64];
    __shared__ float sU[64];
    __shared__ float sNum[16];
    __shared__ float sDen;

    const int f   = blockIdx.x;
    const int tid = threadIdx.x;

    if (tid < 64) {
        sAf[tid] = feat_part[(size_t)f * 64 + tid];
        sU[tid]  = W_u[tid];
    }
    if (tid < 16) sNum[tid] = 0.0f;
    if (tid == 0) sDen = 0.0f;
    __syncthreads();

    float num[16];
#pragma unroll
    for (int e = 0; e < 16; ++e) num[e] = 0.0f;
    float den = 0.0f;

    for (int w = tid; w < W; w += blockDim.x) {
        const float4* hp = (const float4*)(hid_part + (size_t)w * 64);
        float s = 0.0f;
#pragma unroll
        for (int c = 0; c < 16; ++c) {
            const float4 h4 = hp[c];
            const int h = c * 4;
            s = fmaf(FAST_TANH(sAf[h + 0] + h4.x), sU[h + 0], s);
            s = fmaf(FAST_TANH(sAf[h + 1] + h4.y), sU[h + 1], s);
            s = fmaf(FAST_TANH(sAf[h + 2] + h4.z), sU[h + 2], s);
            s = fmaf(FAST_TANH(sAf[h + 3] + h4.w), sU[h + 3], s);
        }
        const float p = mask[(size_t)f * W + w] ? __expf(s) : 0.0f;
        den += p;
        const float4* he = (const float4*)(hidden_emb + (size_t)w * 16);
#pragma unroll
        for (int c = 0; c < 4; ++c) {
            const float4 e4 = he[c];
            num[c * 4 + 0] = fmaf(p, e4.x, num[c * 4 + 0]);
            num[c * 4 + 1] = fmaf(p, e4.y, num[c * 4 + 1]);
            num[c * 4 + 2] = fmaf(p, e4.z, num[c * 4 + 2]);
            num[c * 4 + 3] = fmaf(p, e4.w, num[c * 4 + 3]);
        }
    }

    // wave32 intra-wave reduction
#pragma unroll
    for (int off = 16; off > 0; off >>= 1) {
        den += __shfl_xor(den, off, 32);
#pragma unroll
        for (int e = 0; e < 16; ++e) num[e] += __shfl_xor(num[e], off, 32);
    }
    if ((tid & 31) == 0) {
        atomicAdd(&sDen, den);
#pragma unroll
        for (int e = 0; e < 16; ++e) atomicAdd(&sNum[e], num[e]);
    }
    __syncthreads();

    if (tid < 16) context[(size_t)f * 16 + tid] = sNum[tid] / sDen;
}

// ---------------------------------------------------------------------------
// Kernel C: out[256,16] = values[256,2000] @ context[2000,16]
// 16 M-tiles, one wave each; K loop of 500 V_WMMA_F32_16X16X4_F32 steps.
// ---------------------------------------------------------------------------
__global__ void wmma_out_kernel(const float* __restrict__ values,  // [Bv,F]
                                const float* __restrict__ context, // [F,16]
                                float* __restrict__ out,           // [Bv,16]
                                int F)
{
    const int mt   = blockIdx.x;
    const int lane = threadIdx.x;
    const int half = lane >> 4;
    const int l16  = lane & 15;
    const int row  = mt * 16 + l16;

    v8f acc = {};
    for (int kk = 0; kk < F; kk += 4) {
        const int k = kk + half * 2;
        v2f a, b;
        a.x = values[(size_t)row * F + k];
        a.y = values[(size_t)row * F + k + 1];
        b.x = context[(size_t)k * 16 + l16];
        b.y = context[(size_t)(k + 1) * 16 + l16];
        acc = __builtin_amdgcn_wmma_f32_16x16x4_f32(
            false, a, false, b, (short)0, acc, false, false);
    }
#pragma unroll
    for (int r = 0; r < 8; ++r) {
        const int mrow = mt * 16 + r + half * 8;
        out[(size_t)mrow * 16 + l16] = acc[r];
    }
}

// ---------------------------------------------------------------------------
// Launch: inputs are (values, feature_emb, hidden_emb, W_w, b_w, W_u, mask)
// ---------------------------------------------------------------------------
extern "C" void kernel_launch(void* const* d_in, const int* in_sizes, int n_in,
                              void* d_out, int out_size, void* d_ws, size_t ws_size,
                              hipStream_t stream) {
    (void)in_sizes; (void)n_in; (void)out_size; (void)ws_size;

    constexpr int F = 2000, W = 4000, E = 16, H = 64;

    const float* values      = (const float*)d_in[0]; // [256, F]
    const float* feature_emb = (const float*)d_in[1]; // [F, E]
    const float* hidden_emb  = (const float*)d_in[2]; // [W, E]
    const float* W_w         = (const float*)d_in[3]; // [2E, H]
    const float* b_w         = (const float*)d_in[4]; // [H]
    const float* W_u         = (const float*)d_in[5]; // [H, 1]
    const unsigned char* mask = (const unsigned char*)d_in[6]; // [F, W] bool
    float* out = (float*)d_out;                       // [256, E]

    char* ws = (char*)d_ws;
    float* feat_part = (float*)(ws);                         // [F, H]  512 KB
    float* hid_part  = (float*)(ws + (size_t)F * H * 4);     // [W, H]  1 MB
    float* context   = (float*)(ws + (size_t)(F + W) * H * 4); // [F, E] 128 KB

    // feat_part = feature_emb @ W_w[:E]
    wmma_proj_kernel<<<dim3((F / 16) * 4), dim3(32), 0, stream>>>(
        feature_emb, W_w, nullptr, feat_part, F);
    // hid_part = hidden_emb @ W_w[E:] + b_w
    wmma_proj_kernel<<<dim3((W / 16) * 4), dim3(32), 0, stream>>>(
        hidden_emb, W_w + (size_t)E * H, b_w, hid_part, W);
    // fused tanh + masked softmax over W + context accumulation
    attn_context_kernel<<<dim3(F), dim3(256), 0, stream>>>(
        feat_part, hid_part, W_u, hidden_emb, mask, context, W);
    // out = values @ context
    wmma_out_kernel<<<dim3(256 / 16), dim3(32), 0, stream>>>(
        values, context, out, F);
}